// BasicModel_87935160418316
// MI455X (gfx1250) — compile-verified
//
#include <hip/hip_runtime.h>
#include <hip/hip_bf16.h>
#include <math.h>

// ---------------- geometry constants (match reference) ----------------
#define NB   32
#define NPHI 720
#define NT   512
#define NH   256
#define NW   256

#define RHO_F   1.41421356237309515f          // sqrt(2)
#define DT_F    (RHO_F / 256.0f)              // 2*RHO/T
#define T0_F    (-RHO_F + 0.5f * DT_F)
#define DX_F    0.0078125f                    // 2/256
#define DPHI_F  0.00436332312998582394f       // pi/720

typedef __attribute__((ext_vector_type(2))) float v2f;
typedef __attribute__((ext_vector_type(8))) float v8f;

// ---------------------------------------------------------------------
// Step 1: h = irfft(kernel, n=512).  h[n] = (1/T)*(K0 + 2*sum K_k cos(2pi k n/T)
//                                               + K_{T/2} * (-1)^n)
// ---------------------------------------------------------------------
__global__ __launch_bounds__(256) void fbp_make_h(const float* __restrict__ ker,
                                                  float* __restrict__ h) {
    int n = blockIdx.x * blockDim.x + threadIdx.x;
    if (n >= NT) return;
    const float w0 = 6.28318530717958647692f / (float)NT;   // 2*pi/512
    float s = ker[0];
    #pragma unroll 4
    for (int k = 1; k < NT / 2; ++k) {
        int nk = (n * k) & (NT - 1);         // keep cos argument in [0, 2pi)
        s += 2.0f * ker[k] * cosf(w0 * (float)nk);
    }
    s += ker[NT / 2] * ((n & 1) ? -1.0f : 1.0f);   // cos(pi*n)
    h[n] = s * (1.0f / (float)NT);
}

// ---------------------------------------------------------------------
// Step 2: circulant matrix  C[s][t] = h[(t - s) mod 512]
// ---------------------------------------------------------------------
__global__ __launch_bounds__(256) void fbp_make_circ(const float* __restrict__ h,
                                                     float* __restrict__ C) {
    int idx = blockIdx.x * blockDim.x + threadIdx.x;   // 512*512 total
    int s = idx >> 9;
    int t = idx & (NT - 1);
    C[idx] = h[(t - s) & (NT - 1)];
}

// ---------------------------------------------------------------------
// Step 3: trig table, ang = (p + 0.5) * DPHI
// ---------------------------------------------------------------------
__global__ __launch_bounds__(256) void fbp_make_trig(float2* __restrict__ trig) {
    int p = blockIdx.x * blockDim.x + threadIdx.x;
    if (p >= NPHI) return;
    float ang = ((float)p + 0.5f) * DPHI_F;
    float2 cs;
    __sincosf(ang, &cs.y, &cs.x);   // cs.x = cos, cs.y = sin
    trig[p] = cs;
}

// ---------------------------------------------------------------------
// Step 4: filtered = sino @ C via V_WMMA_F32_16X16X4_F32.
// One wave = one 16x16 output tile, K swept 512 in steps of 4.
//   A (16x4 f32):  lane L, VGPR v  ->  A[L%16][k + 2*(L/16) + v]
//   B (4x16 f32):  lane L, VGPR v  ->  B[k + 2*(L/16) + v][L%16]
//   C/D (16x16):   VGPR v, lanes 0-15 -> row v, lanes 16-31 -> row v+8
// ---------------------------------------------------------------------
__global__ __launch_bounds__(256) void fbp_gemm_wmma(const float* __restrict__ A,
                                                     const float* __restrict__ Bm,
                                                     float* __restrict__ Out) {
    const int lane = threadIdx.x & 31;
    const int wave = threadIdx.x >> 5;
    const int tile = blockIdx.x * 8 + wave;          // 46080 tiles total
    const int tm = (tile >> 5) << 4;                 // row base  (1440 tiles)
    const int tn = (tile & 31) << 4;                 // col base  (32 tiles)

    const int lo    = lane & 15;
    const int khalf = (lane >> 4) << 1;              // 0 or 2

    const float* aptr = A  + (tm + lo) * NT + khalf; // A[m][khalf + k]
    const float* bptr = Bm + khalf * NT + (tn + lo); // B[khalf + k][n]

    v8f acc = {};
    #pragma unroll 8
    for (int k = 0; k < NT; k += 4) {
        const v2f a = *(const v2f*)(aptr + k);       // global_load_b64
        v2f b;
        b.x = bptr[k * NT];
        b.y = bptr[k * NT + NT];
        acc = __builtin_amdgcn_wmma_f32_16x16x4_f32(
            /*neg_a=*/false, a, /*neg_b=*/false, b,
            /*c_mod=*/(short)0, acc, /*reuse_a=*/false, /*reuse_b=*/false);
    }

    const int mhi = (lane >> 4) << 3;                // 0 or 8
    float* op = Out + (tm + mhi) * NT + tn + lo;
    #pragma unroll
    for (int v = 0; v < 8; ++v)
        op[v * NT] = acc[v];
}

// ---------------------------------------------------------------------
// Step 5: backprojection. One thread per output pixel, loop over angles.
// Filtered sinogram (47 MB) is L2-resident; trig staged in LDS.
// out[b][ix][iy] = DPHI * sum_phi lerp(F[b][phi][.], t = x*cos + y*sin)
// ---------------------------------------------------------------------
__global__ __launch_bounds__(256) void fbp_backproject(const float* __restrict__ F,
                                                       const float2* __restrict__ trig,
                                                       float* __restrict__ out) {
    __shared__ float2 s_trig[NPHI];
    for (int p = threadIdx.x; p < NPHI; p += 256)
        s_trig[p] = trig[p];
    __syncthreads();

    const int gid = blockIdx.x * 256 + threadIdx.x;  // 32*256*256 threads
    const int iy = gid & (NW - 1);
    const int ix = (gid >> 8) & (NH - 1);
    const int b  = gid >> 16;

    const float x = -1.0f + ((float)ix + 0.5f) * DX_F;
    const float y = -1.0f + ((float)iy + 0.5f) * DX_F;
    const float invDT = 1.0f / DT_F;
    const float u0 = -T0_F * invDT;                  // u = t/DT - T0/DT

    const float* __restrict__ Fb = F + (size_t)b * (NPHI * NT);
    float acc = 0.0f;

    #pragma unroll 4
    for (int p = 0; p < NPHI; ++p) {
        const float2 cs = s_trig[p];
        const float t = x * cs.x + y * cs.y;
        const float u = fmaf(t, invDT, u0);
        const float fi = floorf(u);
        const float w  = u - fi;
        const int i0 = (int)fi;
        const int i1 = i0 + 1;
        const float w0 = (i0 >= 0 && i0 < NT) ? (1.0f - w) : 0.0f;
        const float w1 = (i1 >= 0 && i1 < NT) ? w : 0.0f;
        const int i0c = min(max(i0, 0), NT - 1);
        const int i1c = min(max(i1, 0), NT - 1);
        const float* row = Fb + p * NT;
        acc = fmaf(row[i0c], w0, acc);
        acc = fmaf(row[i1c], w1, acc);
    }
    out[gid] = acc * DPHI_F;
}

// ---------------------------------------------------------------------
extern "C" void kernel_launch(void* const* d_in, const int* in_sizes, int n_in,
                              void* d_out, int out_size, void* d_ws, size_t ws_size,
                              hipStream_t stream) {
    (void)in_sizes; (void)n_in; (void)out_size; (void)ws_size;

    const float* sinos = (const float*)d_in[0];   // [32, 720, 512] f32
    const float* ker   = (const float*)d_in[1];   // [257] f32

    float* ws   = (float*)d_ws;
    float* h    = ws;                              // 512 floats
    float* Cmat = ws + 512;                        // 512*512 floats (1 MB)
    float* trig = Cmat + NT * NT;                  // 720 float2 (8B aligned)
    float* filt = trig + 2 * NPHI;                 // 23040*512 floats (47 MB)

    fbp_make_h   <<<2,    256, 0, stream>>>(ker, h);
    fbp_make_circ<<<1024, 256, 0, stream>>>(h, Cmat);
    fbp_make_trig<<<3,    256, 0, stream>>>((float2*)trig);

    // 23040/16 = 1440 row-tiles, 512/16 = 32 col-tiles -> 46080 waves, 8/block
    fbp_gemm_wmma<<<46080 / 8, 256, 0, stream>>>(sinos, Cmat, filt);

    // 32*256*256 pixels / 256 threads
    fbp_backproject<<<(NB * NH * NW) / 256, 256, 0, stream>>>(
        filt, (const float2*)trig, (float*)d_out);
}